// CorrVolume1DBlock_283467842296
// MI455X (gfx1250) — compile-verified
//
#include <hip/hip_runtime.h>

typedef __attribute__((ext_vector_type(2))) float v2f;
typedef __attribute__((ext_vector_type(8))) float v8f;

#define BB 8
#define CC 128
#define HH 128
#define WW 256
#define DD 64
#define ND 65          // D+1 disparity slices
#define CK 16          // C-chunk staged in LDS per iteration
#define LS 20          // LDS row stride (floats): even (b64-aligned), conflict-free frag loads
#define PADR 64        // zero rows for w' in [-64, 0): q<0 tiles become 0 without branches

// One workgroup per (n, h): banded Gram G[w,w'] = sum_c x1[c,w]*x2[c,w'] via
// V_WMMA_F32_16X16X4_F32. out[i,w] = G[w, w-i]/C; w-i<0 zeros fall out of the
// zero-padded s2 rows (no branches anywhere in the hot loop).
__global__ __launch_bounds__(512) void corrvol_wmma_f32(
    const float* __restrict__ x1,
    const float* __restrict__ x2,
    float* __restrict__ out)
{
    __shared__ float s1[WW * LS];            // x1 chunk, transposed: s1[w*LS + k]
    __shared__ float s2[(WW + PADR) * LS];   // x2 chunk, rows shifted by +64; rows 0..63 stay zero

    const int lane = threadIdx.x;                        // 0..31
    const int wru  = __builtin_amdgcn_readfirstlane(threadIdx.y); // wave id (scalar)
    const int tid  = wru * 32 + lane;                    // 0..511
    const int hb   = blockIdx.x;                         // 0..127
    const int nb   = blockIdx.y;                         // 0..7

    const int nn = lane & 15;   // N index (and M index for A frag)
    const int hi = lane >> 4;   // half-wave select

    // ---- zero the pad rows of s2 once (w' < 0 region) ----
    for (int e = tid; e < PADR * LS; e += 512) s2[e] = 0.0f;

    // ---- thread-constant fill coordinates: w fixed per thread, k = k0 + 2*jj ----
    const int    w      = tid & (WW - 1);
    const int    k0     = tid >> 8;                      // 0 or 1
    const size_t cplane = (size_t)HH * WW;
    const size_t base   = ((size_t)nb * CC * HH + (size_t)hb) * WW;

    const float* p1 = x1 + base + (size_t)k0 * cplane + (size_t)w;
    const float* p2 = x2 + base + (size_t)k0 * cplane + (size_t)w;
    float* q1 = &s1[w * LS + k0];
    float* q2 = &s2[(w + PADR) * LS + k0];

    // ---- LDS read bases (immediate offsets only in the hot loop) ----
    const float* arow = &s1[(16 * wru + nn) * LS];                       // A: rows of this w-block
    const float* brow = &s2[(16 * wru + nn) * LS];                       // B(dq): +(4-dq)*16*LS from here
    // brow points at row (16*(wru-4) + nn + PADR) since PADR == 64 == 4*16.

    v8f acc[5];
    #pragma unroll
    for (int d = 0; d < 5; ++d) acc[d] = (v8f){};

    // ---- prologue: stage chunk 0 into registers ----
    float r1[CK / 2], r2[CK / 2];
    #pragma unroll
    for (int jj = 0; jj < CK / 2; ++jj) {
        r1[jj] = p1[(size_t)(2 * jj) * cplane];
        r2[jj] = p2[(size_t)(2 * jj) * cplane];
    }

    for (int c0 = 0; c0 < CC; c0 += CK) {
        // ---- commit staged registers to LDS ----
        #pragma unroll
        for (int jj = 0; jj < CK / 2; ++jj) {
            q1[2 * jj] = r1[jj];
            q2[2 * jj] = r2[jj];
        }
        __syncthreads();

        // ---- prefetch next chunk into registers (overlaps with WMMA below) ----
        if (c0 + CK < CC) {
            p1 += (size_t)CK * cplane;
            p2 += (size_t)CK * cplane;
            #pragma unroll
            for (int jj = 0; jj < CK / 2; ++jj) {
                r1[jj] = p1[(size_t)(2 * jj) * cplane];
                r2[jj] = p2[(size_t)(2 * jj) * cplane];
            }
        }

        // ---- branch-free band GEMM: 4 K-steps x 5 tiles ----
        #pragma unroll
        for (int kb = 0; kb < CK; kb += 4) {
            const int koff = kb + hi * 2;                // upper half-wave holds K+2,K+3
            const v2f a = *(const v2f*)&arow[koff];
            v2f bfr[5];
            #pragma unroll
            for (int dq = 0; dq < 5; ++dq)               // dq=4 -> offset 0; dq=0 -> +4*16*LS
                bfr[dq] = *(const v2f*)&brow[(size_t)(4 - dq) * 16 * LS + koff];
            #pragma unroll
            for (int dq = 0; dq < 5; ++dq)
                acc[dq] = __builtin_amdgcn_wmma_f32_16x16x4_f32(
                    false, a, false, bfr[dq], (short)0, acc[dq], false, false);
        }
        __syncthreads();
    }

    // ---- store: D element (M=v+8*hi -> w, N=nn -> w'), i = w - w' ----
    const float  scale = 1.0f / (float)CC;
    const long   HW    = (long)HH * WW;
    // index(dq,v) = L0 + dq*16*HW + v*(HW + 1); i(dq,v) = 16*dq + v + 8*hi - nn
    const long L0 = (long)nb * ND * HH * WW + (long)hb * WW
                  + (long)(8 * hi - nn) * HW + (long)(16 * wru + 8 * hi);
    #pragma unroll
    for (int dq = 0; dq < 5; ++dq) {
        #pragma unroll
        for (int v = 0; v < 8; ++v) {
            const int i = 16 * dq + v + 8 * hi - nn;     // disparity for this element
            if (i >= 0 && i <= DD) {
                out[L0 + (long)dq * 16 * HW + (long)v * (HW + 1)] = acc[dq][v] * scale;
            }
        }
    }
}

extern "C" void kernel_launch(void* const* d_in, const int* in_sizes, int n_in,
                              void* d_out, int out_size, void* d_ws, size_t ws_size,
                              hipStream_t stream) {
    const float* x1 = (const float*)d_in[0];
    const float* x2 = (const float*)d_in[1];
    float* out = (float*)d_out;

    dim3 grid(HH, BB);     // one workgroup per (h, n)
    dim3 block(32, 16);    // 16 wave32s
    hipLaunchKernelGGL(corrvol_wmma_f32, grid, block, 0, stream, x1, x2, out);
}